// DTLN_RK_P2_53429393162682
// MI455X (gfx1250) — compile-verified
//
#include <hip/hip_runtime.h>
#include <hip/hip_bf16.h>

#define FRAME 1024
#define ENC   256
#define HID   128
#define GATES (4 * HID)   // 512

typedef float v2f __attribute__((ext_vector_type(2)));
typedef float v8f __attribute__((ext_vector_type(8)));

__device__ __forceinline__ float sigmoidf_(float x) {
    return 1.0f / (1.0f + expf(-x));
}

// Non-blocking streaming prefetch of a weight matrix into the cache
// hierarchy (lowers to global_prefetch_b8; does not touch LOADcnt).
// 128-byte cacheline stride, strided across all threads of the block.
__device__ __forceinline__ void prefetch_weights(const float* __restrict__ p,
                                                 size_t bytes, int t, int nthreads) {
    const char* base = (const char*)p;
    for (size_t off = (size_t)t * 128; off < bytes; off += (size_t)nthreads * 128) {
        __builtin_prefetch(base + off, 0, 3);
    }
}

// Accumulate a 16-row matvec block with V_WMMA_F32_16X16X4_F32.
// Wrow: pointer to W[row0 + (lane&15)] row start (already includes column base)
// x:    input vector (LDS), read so that every B column replicates x -> all 16
//       D columns hold the matvec result.
// koff: 2*(lane>>4) per the 32-bit A/B VGPR layouts (K pair per half-wave)
__device__ __forceinline__ v8f mv_acc(const float* __restrict__ Wrow,
                                      const float* __restrict__ x,
                                      int K, int koff, v8f c) {
#pragma unroll 4
    for (int k = 0; k < K; k += 4) {
        const int ka = k + koff;                       // even -> 8B aligned
        v2f a = *(const v2f*)(Wrow + ka);              // A: 16x4 f32 tile slice
        v2f b = *(const v2f*)(x + ka);                 // B: x replicated per column
        c = __builtin_amdgcn_wmma_f32_16x16x4_f32(
                /*neg_a=*/false, a, /*neg_b=*/false, b,
                /*c_mod=*/(short)0, c, /*reuse_a=*/false, /*reuse_b=*/false);
    }
    return c;
}

__global__ __launch_bounds__(1024)
void dtln_fused_kernel(const float* __restrict__ y1,
                       const float* __restrict__ h1_in, const float* __restrict__ c1_in,
                       const float* __restrict__ h2_in, const float* __restrict__ c2_in,
                       const float* __restrict__ enc_W,
                       const float* __restrict__ gamma, const float* __restrict__ beta,
                       const float* __restrict__ Wih1, const float* __restrict__ Whh1,
                       const float* __restrict__ bih1, const float* __restrict__ bhh1,
                       const float* __restrict__ Wih2, const float* __restrict__ Whh2,
                       const float* __restrict__ bih2, const float* __restrict__ bhh2,
                       const float* __restrict__ dense_W, const float* __restrict__ dense_b,
                       const float* __restrict__ dec_W,
                       float* __restrict__ out) {
    __shared__ __align__(16) float y_s[FRAME];
    __shared__ __align__(16) float enc_s[ENC];
    __shared__ __align__(16) float nrm_s[ENC];
    __shared__ __align__(16) float red_s[ENC];
    __shared__ __align__(16) float z_s[GATES];
    __shared__ __align__(16) float h1_s[HID];
    __shared__ __align__(16) float c1_s[HID];
    __shared__ __align__(16) float h2_s[HID];
    __shared__ __align__(16) float c2_s[HID];
    __shared__ __align__(16) float est_s[ENC];

    const int t    = threadIdx.x;        // 0..1023, 32 waves
    const int lane = t & 31;
    const int wave = t >> 5;
    const int m    = lane & 15;          // A-matrix row within 16-row block
    const int koff = (lane >> 4) << 1;   // K sub-offset {0,2} per half-wave

    // ---- stage 0: stage inputs into LDS, zero encoder accumulator ----
    y_s[t] = y1[t];
    if (t < ENC) enc_s[t] = 0.0f;
    if (t < HID) {
        h1_s[t] = h1_in[t];  c1_s[t] = c1_in[t];
        h2_s[t] = h2_in[t];  c2_s[t] = c2_in[t];
    }

    // Kick off non-blocking prefetch of every later stage's weights so their
    // HBM->L2->WGP$ latency overlaps the encoder WMMA loop (serial-chain
    // latency is the dominant cost; compute & bandwidth are nowhere near peak).
    prefetch_weights(Wih1,    (size_t)GATES * ENC  * sizeof(float), t, 1024);
    prefetch_weights(Whh1,    (size_t)GATES * HID  * sizeof(float), t, 1024);
    prefetch_weights(Wih2,    (size_t)GATES * HID  * sizeof(float), t, 1024);
    prefetch_weights(Whh2,    (size_t)GATES * HID  * sizeof(float), t, 1024);
    prefetch_weights(dense_W, (size_t)ENC   * HID  * sizeof(float), t, 1024);
    prefetch_weights(dec_W,   (size_t)FRAME * ENC  * sizeof(float), t, 1024);

    __syncthreads();

    // ---- stage 1: encoder enc = enc_W[256,1024] @ y  (split-K, 2 waves/block) ----
    {
        const int rb = wave >> 1;         // 16 row blocks
        const int kh = wave & 1;          // K half: [0,512) or [512,1024)
        const float* Wrow = enc_W + (size_t)(rb * 16 + m) * FRAME + kh * 512;
        v8f c = {0.f,0.f,0.f,0.f,0.f,0.f,0.f,0.f};
        c = mv_acc(Wrow, y_s + kh * 512, 512, koff, c);
        if (m == 0) {                     // lanes 0 and 16: columns N==0
            const int base = rb * 16 + ((lane >> 4) << 3);
#pragma unroll
            for (int r = 0; r < 8; ++r) atomicAdd(&enc_s[base + r], c[r]);
        }
    }
    __syncthreads();

    // ---- stage 2: global-statistic instant layer norm over all 256 elems ----
    if (t < ENC) red_s[t] = enc_s[t];
    __syncthreads();
    for (int s = ENC / 2; s > 0; s >>= 1) {
        if (t < s) red_s[t] += red_s[t + s];
        __syncthreads();
    }
    const float mean = red_s[0] * (1.0f / ENC);
    __syncthreads();
    if (t < ENC) { float d = enc_s[t] - mean; red_s[t] = d * d; }
    __syncthreads();
    for (int s = ENC / 2; s > 0; s >>= 1) {
        if (t < s) red_s[t] += red_s[t + s];
        __syncthreads();
    }
    const float rstd = rsqrtf(red_s[0] * (1.0f / ENC) + 1e-7f);
    if (t < ENC) nrm_s[t] = (enc_s[t] - mean) * rstd * gamma[t] + beta[t];
    __syncthreads();

    // ---- stage 3: LSTM1 z = Wih1@nrm + Whh1@h1 + bih1 + bhh1 (32 blocks, 1/wave) ----
    {
        const int row0 = wave * 16;
        v8f c = {0.f,0.f,0.f,0.f,0.f,0.f,0.f,0.f};
        c = mv_acc(Wih1 + (size_t)(row0 + m) * ENC, nrm_s, ENC, koff, c);
        c = mv_acc(Whh1 + (size_t)(row0 + m) * HID, h1_s, HID, koff, c);
        if (m == 0) {
            const int base = row0 + ((lane >> 4) << 3);
#pragma unroll
            for (int r = 0; r < 8; ++r)
                z_s[base + r] = c[r] + bih1[base + r] + bhh1[base + r];
        }
    }
    __syncthreads();
    if (t < HID) {
        const float ig = sigmoidf_(z_s[t]);
        const float fg = sigmoidf_(z_s[HID + t]);
        const float gg = tanhf(z_s[2 * HID + t]);
        const float og = sigmoidf_(z_s[3 * HID + t]);
        const float cn = fg * c1_s[t] + ig * gg;
        c1_s[t] = cn;
        h1_s[t] = og * tanhf(cn);
    }
    __syncthreads();

    // ---- stage 4: LSTM2 z = Wih2@h1 + Whh2@h2 + bih2 + bhh2 ----
    {
        const int row0 = wave * 16;
        v8f c = {0.f,0.f,0.f,0.f,0.f,0.f,0.f,0.f};
        c = mv_acc(Wih2 + (size_t)(row0 + m) * HID, h1_s, HID, koff, c);
        c = mv_acc(Whh2 + (size_t)(row0 + m) * HID, h2_s, HID, koff, c);
        if (m == 0) {
            const int base = row0 + ((lane >> 4) << 3);
#pragma unroll
            for (int r = 0; r < 8; ++r)
                z_s[base + r] = c[r] + bih2[base + r] + bhh2[base + r];
        }
    }
    __syncthreads();
    if (t < HID) {
        const float ig = sigmoidf_(z_s[t]);
        const float fg = sigmoidf_(z_s[HID + t]);
        const float gg = tanhf(z_s[2 * HID + t]);
        const float og = sigmoidf_(z_s[3 * HID + t]);
        const float cn = fg * c2_s[t] + ig * gg;
        c2_s[t] = cn;
        h2_s[t] = og * tanhf(cn);
    }
    __syncthreads();

    // ---- stage 5: mask = sigmoid(dense_W@h2 + b); estimated = mask * enc ----
    if (wave < 16) {
        const int row0 = wave * 16;
        v8f c = {0.f,0.f,0.f,0.f,0.f,0.f,0.f,0.f};
        c = mv_acc(dense_W + (size_t)(row0 + m) * HID, h2_s, HID, koff, c);
        if (m == 0) {
            const int base = row0 + ((lane >> 4) << 3);
#pragma unroll
            for (int r = 0; r < 8; ++r)
                z_s[base + r] = c[r] + dense_b[base + r];
        }
    }
    __syncthreads();
    if (t < ENC) est_s[t] = sigmoidf_(z_s[t]) * enc_s[t];
    __syncthreads();

    // ---- stage 6: decoder out[0:1024] = dec_W[1024,256] @ estimated ----
    for (int blk = wave; blk < 64; blk += 32) {     // 2 blocks per wave, uniform
        const int row0 = blk * 16;
        v8f c = {0.f,0.f,0.f,0.f,0.f,0.f,0.f,0.f};
        c = mv_acc(dec_W + (size_t)(row0 + m) * ENC, est_s, ENC, koff, c);
        if (m == 0) {
            const int base = row0 + ((lane >> 4) << 3);
#pragma unroll
            for (int r = 0; r < 8; ++r) out[base + r] = c[r];
        }
    }

    // ---- stage 7: state outputs (h1, c1, h2, c2) ----
    if (t < HID) {
        out[FRAME + t]            = h1_s[t];
        out[FRAME + HID + t]      = c1_s[t];
        out[FRAME + 2 * HID + t]  = h2_s[t];
        out[FRAME + 3 * HID + t]  = c2_s[t];
    }
}

extern "C" void kernel_launch(void* const* d_in, const int* in_sizes, int n_in,
                              void* d_out, int out_size, void* d_ws, size_t ws_size,
                              hipStream_t stream) {
    (void)in_sizes; (void)n_in; (void)out_size; (void)d_ws; (void)ws_size;
    const float* y1      = (const float*)d_in[0];
    const float* h1_in   = (const float*)d_in[1];
    const float* c1_in   = (const float*)d_in[2];
    const float* h2_in   = (const float*)d_in[3];
    const float* c2_in   = (const float*)d_in[4];
    const float* enc_W   = (const float*)d_in[5];
    const float* gamma   = (const float*)d_in[6];
    const float* beta    = (const float*)d_in[7];
    const float* Wih1    = (const float*)d_in[8];
    const float* Whh1    = (const float*)d_in[9];
    const float* bih1    = (const float*)d_in[10];
    const float* bhh1    = (const float*)d_in[11];
    const float* Wih2    = (const float*)d_in[12];
    const float* Whh2    = (const float*)d_in[13];
    const float* bih2    = (const float*)d_in[14];
    const float* bhh2    = (const float*)d_in[15];
    const float* dense_W = (const float*)d_in[16];
    const float* dense_b = (const float*)d_in[17];
    const float* dec_W   = (const float*)d_in[18];
    float* out = (float*)d_out;

    dtln_fused_kernel<<<1, 1024, 0, stream>>>(
        y1, h1_in, c1_in, h2_in, c2_in, enc_W, gamma, beta,
        Wih1, Whh1, bih1, bhh1, Wih2, Whh2, bih2, bhh2,
        dense_W, dense_b, dec_W, out);
}